// MeshLoss_15857019256896
// MI455X (gfx1250) — compile-verified
//
#include <hip/hip_runtime.h>

typedef __attribute__((ext_vector_type(2))) float v2f;
typedef __attribute__((ext_vector_type(8))) float v8f;

#define NPTS   5329          // 73*73 refined top points per batch
#define NPAD   5376          // 84 * 64 (4 tiles of 16 per wave)
#define NGRP   84            // row-groups of 64
#define TPW    4             // 16-row tiles per wave
#define MPTS   8192
#define MTILES 512
#define MSPLIT 4             // M-dimension splits (waves per row-group)
#define MT_PER (MTILES / MSPLIT)
#define FEM_N  18750         // 2*3*25*5*25

// Monotone float -> uint key (order preserving)
__device__ __forceinline__ unsigned fkey(float f) {
    unsigned u = __float_as_uint(f);
    return (u & 0x80000000u) ? ~u : (u | 0x80000000u);
}
__device__ __forceinline__ float fdec(unsigned k) {
    unsigned u = (k & 0x80000000u) ? (k & 0x7FFFFFFFu) : ~k;
    return __uint_as_float(u);
}

// ---------------------------------------------------------------------------
// Kernel 1: build A quads (refined top, [-2x,-2y,-2z,|x|^2]), B quads
// ([x,y,z,1]), pc norms, and init both atomic-min key arrays.
// ---------------------------------------------------------------------------
__global__ __launch_bounds__(256) void prep_kernel(const float* __restrict__ nm,
                                                   const float* __restrict__ pc,
                                                   float* __restrict__ topq,
                                                   float* __restrict__ pcq,
                                                   float* __restrict__ pcnorm,
                                                   unsigned* __restrict__ d2keys,
                                                   unsigned* __restrict__ d1keys) {
    int tid = blockIdx.x * 256 + threadIdx.x;

    if (tid < 2 * MPTS) {
        int b = tid >> 13, m = tid & (MPTS - 1);
        float x = pc[(b * 3 + 0) * MPTS + m];
        float y = pc[(b * 3 + 1) * MPTS + m];
        float z = pc[(b * 3 + 2) * MPTS + m];
        ((float4*)pcq)[(size_t)b * MPTS + m] = make_float4(x, y, z, 1.0f);
        pcnorm[tid] = x * x + y * y + z * z;
        d2keys[tid] = 0xFFFFFFFFu;
    }

    if (tid < 2 * NPTS) d1keys[tid] = 0xFFFFFFFFu;

    if (tid < 2 * NPAD) {
        int b = tid / NPAD, n = tid % NPAD;
        float p0, p1, p2;
        if (n < NPTS) {
            int i = n / 73, j = n % 73;
            int ri = i / 3, ci = j / 3;
            int ri2 = min(ri + 1, 24), ci2 = min(ci + 1, 24);
            float wi = (float)(i - ri * 3) * (1.0f / 3.0f);
            float wj = (float)(j - ci * 3) * (1.0f / 3.0f);
            float p[3];
            #pragma unroll
            for (int d = 0; d < 3; ++d) {
                // network_mesh[b][d][r][4][c]; separable linear == bilinear
                const float* a = nm + (size_t)(b * 3 + d) * 3125 + 100;
                float v00 = a[ri * 125 + ci];
                float v01 = a[ri * 125 + ci2];
                float v10 = a[ri2 * 125 + ci];
                float v11 = a[ri2 * 125 + ci2];
                p[d] = (1.0f - wi) * ((1.0f - wj) * v00 + wj * v01) +
                       wi * ((1.0f - wj) * v10 + wj * v11);
            }
            p0 = p[0]; p1 = p[1]; p2 = p[2];
        } else {
            // Padded rows: huge coords -> huge distances, never win any min.
            p0 = p1 = p2 = 1.0e6f;
        }
        float nrm = p0 * p0 + p1 * p1 + p2 * p2;
        ((float4*)topq)[(size_t)b * NPAD + n] =
            make_float4(-2.0f * p0, -2.0f * p1, -2.0f * p2, nrm);
    }
}

// ---------------------------------------------------------------------------
// Kernel 2: Chamfer GEMM via V_WMMA_F32_16X16X4_F32.
// One wave owns 4 row-tiles (64 rows) and 1/4 of the B tiles:
//   per B tile: 1 B-frag load + 1 norm load + 4 WMMAs, then one
//   column-min shuffle + one atomicMin.  Row mins kept in 4x8 accum regs.
// ---------------------------------------------------------------------------
__global__ __launch_bounds__(32) void chamfer_kernel(const float* __restrict__ topq,
                                                     const float* __restrict__ pcq,
                                                     const float* __restrict__ pcnorm,
                                                     unsigned* __restrict__ d2keys,
                                                     unsigned* __restrict__ d1keys) {
    const int b    = blockIdx.z;
    const int n0   = blockIdx.x * (TPW * 16);
    const int mt0  = blockIdx.y * MT_PER;
    const int lane = threadIdx.x;
    const int half = lane >> 4;    // 0: K={0,1}, 1: K={2,3}
    const int l15  = lane & 15;

    // A fragments (ISA 32-bit 16x4 layout): lane<16 -> (K0,K1) of row l15,
    // lane>=16 -> (K2,K3) of row l15.
    v2f afrag[TPW];
    #pragma unroll
    for (int t = 0; t < TPW; ++t) {
        const float* ta = topq + (((size_t)b * NPAD + n0 + t * 16 + l15) * 4 + 2 * half);
        afrag[t].x = ta[0];
        afrag[t].y = ta[1];
    }

    v8f rmin[TPW];
    #pragma unroll
    for (int t = 0; t < TPW; ++t)
        #pragma unroll
        for (int v = 0; v < 8; ++v) rmin[t][v] = 3.0e38f;

    const float* pq = pcq + (size_t)b * MPTS * 4;
    const float* pn = pcnorm + (size_t)b * MPTS;
    unsigned* keys2 = d2keys + (size_t)b * MPTS;

    for (int mt = mt0; mt < mt0 + MT_PER; ++mt) {
        const int m0 = mt * 16;
        const float* tb = pq + ((size_t)(m0 + l15) * 4 + 2 * half);
        v2f bfrag; bfrag.x = tb[0]; bfrag.y = tb[1];
        const float cn = pn[m0 + l15];

        float cmin = 3.0e38f;
        #pragma unroll
        for (int t = 0; t < TPW; ++t) {
            v8f c = {0.f, 0.f, 0.f, 0.f, 0.f, 0.f, 0.f, 0.f};
            v8f d = __builtin_amdgcn_wmma_f32_16x16x4_f32(
                false, afrag[t], false, bfrag, (short)0, c, false, false);
            #pragma unroll
            for (int v = 0; v < 8; ++v) {
                float val = d[v] + cn;               // full squared distance
                rmin[t][v] = fminf(rmin[t][v], val); // running row-min
                cmin       = fminf(cmin, val);       // column min (64 rows)
            }
        }
        // combine rows held by the two lane-halves, then one atomic / column
        cmin = fminf(cmin, __shfl_xor(cmin, 16, 32));
        if (half == 0)
            atomicMin(&keys2[m0 + l15], fkey(cmin));
    }

    // Row mins: reduce over the 16 columns held by each 16-lane group,
    // then merge across M-splits with keyed atomicMin (deterministic).
    #pragma unroll
    for (int t = 0; t < TPW; ++t) {
        #pragma unroll
        for (int mask = 1; mask < 16; mask <<= 1)
            #pragma unroll
            for (int v = 0; v < 8; ++v)
                rmin[t][v] = fminf(rmin[t][v], __shfl_xor(rmin[t][v], mask, 32));
    }
    if (l15 == 0) {
        #pragma unroll
        for (int t = 0; t < TPW; ++t) {
            const int rbase = n0 + t * 16 + half * 8;
            #pragma unroll
            for (int v = 0; v < 8; ++v) {
                int n = rbase + v;
                if (n < NPTS)
                    atomicMin(&d1keys[(size_t)b * NPTS + n], fkey(rmin[t][v]));
            }
        }
    }
}

// ---------------------------------------------------------------------------
// Kernel 3: deterministic final reduction (fixed-order sums + LDS tree).
// ---------------------------------------------------------------------------
__global__ __launch_bounds__(256) void finalize_kernel(const float* __restrict__ nm,
                                                       const float* __restrict__ fm,
                                                       const unsigned* __restrict__ d1keys,
                                                       const unsigned* __restrict__ d2keys,
                                                       float* __restrict__ out) {
    __shared__ float s1[256], s2[256], s3[256];
    const int t = threadIdx.x;
    float a1 = 0.f, a2 = 0.f, a3 = 0.f;
    for (int i = t; i < 2 * NPTS; i += 256) a1 += fdec(d1keys[i]);
    for (int i = t; i < 2 * MPTS; i += 256) a2 += fdec(d2keys[i]);
    for (int i = t; i < FEM_N; i += 256) {
        float d = nm[i] - fm[i];
        a3 += d * d;
    }
    s1[t] = a1; s2[t] = a2; s3[t] = a3;
    __syncthreads();
    for (int off = 128; off > 0; off >>= 1) {
        if (t < off) { s1[t] += s1[t + off]; s2[t] += s2[t + off]; s3[t] += s3[t + off]; }
        __syncthreads();
    }
    if (t == 0)
        out[0] = s1[0] * (1.0f / (2.0f * NPTS)) +
                 s2[0] * (1.0f / (2.0f * MPTS)) +
                 s3[0] * (1.0f / (float)FEM_N);   // WEIGHT == 1.0
}

// ---------------------------------------------------------------------------
extern "C" void kernel_launch(void* const* d_in, const int* in_sizes, int n_in,
                              void* d_out, int out_size, void* d_ws, size_t ws_size,
                              hipStream_t stream) {
    const float* nm = (const float*)d_in[0];   // network_mesh (2,3,25,5,25)
    const float* pc = (const float*)d_in[1];   // pc (2,3,8192)
    const float* fm = (const float*)d_in[2];   // fem_mesh (2,3,25,5,25)
    float* out = (float*)d_out;

    char* ws = (char*)d_ws;
    float*    topq   = (float*)(ws);                    // 2*5376*4 f32 = 172032 B
    float*    pcq    = (float*)(ws + 172032);           // 2*8192*4 f32 = 262144 B
    float*    pcnorm = (float*)(ws + 434176);           // 2*8192 f32   =  65536 B
    unsigned* d2keys = (unsigned*)(ws + 499712);        // 2*8192 u32   =  65536 B
    unsigned* d1keys = (unsigned*)(ws + 565248);        // 2*5329 u32   =  42632 B

    prep_kernel<<<64, 256, 0, stream>>>(nm, pc, topq, pcq, pcnorm, d2keys, d1keys);
    chamfer_kernel<<<dim3(NGRP, MSPLIT, 2), 32, 0, stream>>>(topq, pcq, pcnorm,
                                                             d2keys, d1keys);
    finalize_kernel<<<1, 256, 0, stream>>>(nm, fm, d1keys, d2keys, out);
}